// AdaCLIP_54623394070583
// MI455X (gfx1250) — compile-verified
//
#include <hip/hip_runtime.h>
#include <hip/hip_bf16.h>

typedef __attribute__((ext_vector_type(2))) float v2f;
typedef __attribute__((ext_vector_type(8))) float v8f;
typedef __attribute__((ext_vector_type(4))) unsigned int u32x4;
typedef __attribute__((ext_vector_type(4))) int i32x4;
typedef __attribute__((ext_vector_type(8))) int i32x8;

#define B_      8
#define L_      1369
#define C_      1024
#define CC_     4096      // 4 layers concatenated
#define KCL     20
#define KPAD    32
#define NSEL    100
#define MPAD    112
#define ITERS   10
#define KSLAB   128                  // channels staged per LDS slab
#define NSLAB   (CC_ / KSLAB)        // 32 slabs
#define NEG_INF (-3.402823466e38f)

#if __has_builtin(__builtin_amdgcn_tensor_load_to_lds) && __has_builtin(__builtin_amdgcn_s_wait_tensorcnt)
#define USE_TDM 1
#else
#define USE_TDM 0
#endif

// ---------------------------------------------------------------------------
// Kernel 1: score -> top-100 -> gather into padded X, init centers C
// ---------------------------------------------------------------------------
__global__ void k_select_gather(const float* __restrict__ pt0, const float* __restrict__ pt1,
                                const float* __restrict__ pt2, const float* __restrict__ pt3,
                                const float* __restrict__ am0, const float* __restrict__ am1,
                                const float* __restrict__ am2, const float* __restrict__ am3,
                                float* __restrict__ Xbuf, float* __restrict__ Cbuf,
                                int* __restrict__ selIdx)
{
    const int b = blockIdx.x;
    const int t = threadIdx.x;

    __shared__ float score[L_];
    __shared__ float bv[256];
    __shared__ int   bi[256];
    __shared__ int   sel[NSEL];

    const float* A0 = am0 + (size_t)b * L_ * 2;
    const float* A1 = am1 + (size_t)b * L_ * 2;
    const float* A2 = am2 + (size_t)b * L_ * 2;
    const float* A3 = am3 + (size_t)b * L_ * 2;

    for (int l = t; l < L_; l += 256) {
        float a0 = 0.25f * (A0[2*l]   + A1[2*l]   + A2[2*l]   + A3[2*l]);
        float a1 = 0.25f * (A0[2*l+1] + A1[2*l+1] + A2[2*l+1] + A3[2*l+1]);
        score[l] = 1.0f / (1.0f + expf(a0 - a1));   // softmax(.,2)[:,1]
    }
    __syncthreads();

    for (int i = 0; i < NSEL; ++i) {
        float best = NEG_INF;
        int   besti = 0x7fffffff;
        for (int l = t; l < L_; l += 256) {
            float v = score[l];
            if (v > best || (v == best && l < besti)) { best = v; besti = l; }
        }
        bv[t] = best; bi[t] = besti;
        __syncthreads();
        for (int s = 128; s > 0; s >>= 1) {
            if (t < s) {
                float ov = bv[t + s]; int oi = bi[t + s];
                if (ov > bv[t] || (ov == bv[t] && oi < bi[t])) { bv[t] = ov; bi[t] = oi; }
            }
            __syncthreads();
        }
        if (t == 0) {
            int w = bi[0];
            sel[i] = w;
            selIdx[b * NSEL + i] = w;
            score[w] = NEG_INF;
        }
        __syncthreads();
    }

    float* Xb = Xbuf + (size_t)b * MPAD * CC_;
    float* Cb = Cbuf + (size_t)b * KPAD * CC_;
    for (int i = 0; i < NSEL; ++i) {
        int row = sel[i];
        for (int c = t; c < CC_; c += 256) {
            int l  = c >> 10;
            int cc = c & 1023;
            const float* p = (l < 2) ? (l == 0 ? pt0 : pt1) : (l == 2 ? pt2 : pt3);
            float v = p[((size_t)b * L_ + row) * C_ + cc];
            Xb[(size_t)i * CC_ + c] = v;
            if (i < KCL) Cb[(size_t)i * CC_ + c] = v;
        }
    }
    for (int i = NSEL; i < MPAD; ++i)
        for (int c = t; c < CC_; c += 256) Xb[(size_t)i * CC_ + c] = 0.0f;
    for (int k = KCL; k < KPAD; ++k)
        for (int c = t; c < CC_; c += 256) Cb[(size_t)k * CC_ + c] = 0.0f;
}

// ---------------------------------------------------------------------------
// TDM: 2D tile load Global -> LDS  (D# per CDNA5 ISA)
//   data_size = 4 bytes, tile (tile_w x tile_h), row stride = row_stride elems
//   This toolchain exposes the 6-arg builtin (probe-verified arity); the three
//   trailing descriptor-group args beyond groups 0/1 are zeros (2D tensor).
// ---------------------------------------------------------------------------
#if USE_TDM
__device__ __forceinline__ void tdm_load_2d(unsigned lds_off_bytes, const float* gptr,
                                            unsigned tile_w, unsigned tile_h,
                                            unsigned tensor_w, unsigned tensor_h,
                                            unsigned row_stride)
{
    unsigned long long ga = (unsigned long long)(const void*)gptr;
    u32x4 g0;
    g0.x = 0x1u;                                             // count=1, user D#
    g0.y = lds_off_bytes;                                    // lds_addr
    g0.z = (unsigned)(ga & 0xFFFFFFFFull);                   // global_addr[31:0]
    g0.w = (unsigned)((ga >> 32) & 0x01FFFFFFull) | (2u << 30); // addr[56:32] | type=2
    i32x8 g1;
    g1[0] = (int)(2u << 16);                                 // data_size = 4B
    g1[1] = (int)((tensor_w & 0xFFFFu) << 16);               // tensor_dim0[15:0]
    g1[2] = (int)(((tensor_w >> 16) & 0xFFFFu) | ((tensor_h & 0xFFFFu) << 16));
    g1[3] = (int)(((tensor_h >> 16) & 0xFFFFu) | ((tile_w & 0xFFFFu) << 16)); // tile_dim0
    g1[4] = (int)(tile_h & 0xFFFFu);                         // tile_dim1 (tile_dim2=0)
    g1[5] = (int)row_stride;                                 // tensor_dim0_stride lo32
    g1[6] = 0;
    g1[7] = 0;
    i32x4 z4 = {0, 0, 0, 0};                                 // 2D: groups 2/3 unused
    i32x8 z8 = {0, 0, 0, 0, 0, 0, 0, 0};
    __builtin_amdgcn_tensor_load_to_lds(g0, g1, z4, z4, z8, 0);
}
#endif

// ---------------------------------------------------------------------------
// Kernel 2: k-means. Dot = X(112x4096) @ C^T(4096x32) via WMMA f32 16x16x4,
// fed from LDS K-slabs staged by the Tensor Data Mover (or coop copy).
// one block per batch, 256 threads = 8 waves, 14 output tiles.
// ---------------------------------------------------------------------------
__global__ void k_kmeans(const float* __restrict__ Xbuf, float* __restrict__ Cbuf,
                         int* __restrict__ labelsG, int* __restrict__ cntG)
{
    const int b    = blockIdx.x;
    const int t    = threadIdx.x;
    const int wave = t >> 5;
    const int lane = t & 31;
    const int half = lane >> 4;
    const int lm   = lane & 15;

    const float* Xb = Xbuf + (size_t)b * MPAD * CC_;
    float*       Cb = Cbuf + (size_t)b * KPAD * CC_;

    __shared__ float dotL[MPAD * KPAD];                // 14 KB
    __shared__ __align__(16) union {
        float sums[KCL * 1024];                        // 80 KB (update phase)
        struct {                                       // 72 KB (assignment phase)
            float Xs[MPAD * KSLAB];
            float Cs[KPAD * KSLAB];
        } st;
    } u;
    __shared__ float cnorm[KCL];
    __shared__ int   labels[MPAD];
    __shared__ int   cnt[KCL];

#if USE_TDM
    const unsigned xs_lds = (unsigned)(unsigned long long)
        (__attribute__((address_space(3))) const float*)u.st.Xs;
    const unsigned cs_lds = (unsigned)(unsigned long long)
        (__attribute__((address_space(3))) const float*)u.st.Cs;
#endif

    const int tile0 = wave;            // 0..7   (always < 14)
    const int tile1 = wave + 8;        // 8..15  (< 14 for waves 0..5)
    const int m0a = (tile0 % 7) * 16, n0a = (tile0 / 7) * 16;
    const int m0b = (tile1 % 7) * 16, n0b = (tile1 / 7) * 16;

    for (int it = 0; it <= ITERS; ++it) {
        // --- |c_k|^2 ---
        if (t < KCL) cnorm[t] = 0.0f;
        __syncthreads();
        for (int k = 0; k < KCL; ++k) {
            float p = 0.0f;
            for (int c = t; c < CC_; c += 256) { float v = Cb[(size_t)k * CC_ + c]; p += v * v; }
            atomicAdd(&cnorm[k], p);
        }

        // --- Dot = X @ C^T, K-slab staged through LDS ---
        v8f acc0 = {0.f,0.f,0.f,0.f,0.f,0.f,0.f,0.f};
        v8f acc1 = {0.f,0.f,0.f,0.f,0.f,0.f,0.f,0.f};

        for (int slab = 0; slab < NSLAB; ++slab) {
            const int ks = slab * KSLAB;
#if USE_TDM
            if (wave == 0) {
                tdm_load_2d(xs_lds, Xb + ks, KSLAB, MPAD, CC_, MPAD, CC_);
                tdm_load_2d(cs_lds, Cb + ks, KSLAB, KPAD, CC_, KPAD, CC_);
                __builtin_amdgcn_s_wait_tensorcnt(0);
            }
#else
            for (int j = t; j < MPAD * (KSLAB / 4); j += 256) {
                int r = j >> 5, q = j & 31;            // KSLAB/4 = 32 float4 per row
                reinterpret_cast<float4*>(u.st.Xs)[r * 32 + q] =
                    reinterpret_cast<const float4*>(Xb + (size_t)r * CC_ + ks)[q];
            }
            for (int j = t; j < KPAD * (KSLAB / 4); j += 256) {
                int r = j >> 5, q = j & 31;
                reinterpret_cast<float4*>(u.st.Cs)[r * 32 + q] =
                    reinterpret_cast<const float4*>(Cb + (size_t)r * CC_ + ks)[q];
            }
#endif
            __syncthreads();

            {   // tile 0 (all waves)
                const float* xa = u.st.Xs + (m0a + lm) * KSLAB + 2 * half;
                const float* ca = u.st.Cs + (n0a + lm) * KSLAB + 2 * half;
#pragma unroll 8
                for (int kk = 0; kk < KSLAB / 4; ++kk) {
                    v2f a  = *reinterpret_cast<const v2f*>(xa + 4 * kk);
                    v2f bb = *reinterpret_cast<const v2f*>(ca + 4 * kk);
                    acc0 = __builtin_amdgcn_wmma_f32_16x16x4_f32(false, a, false, bb,
                                                                 (short)0, acc0, false, false);
                }
            }
            if (tile1 < 14) {   // tile 1 (waves 0..5), wave-uniform branch, EXEC all ones
                const float* xa = u.st.Xs + (m0b + lm) * KSLAB + 2 * half;
                const float* ca = u.st.Cs + (n0b + lm) * KSLAB + 2 * half;
#pragma unroll 8
                for (int kk = 0; kk < KSLAB / 4; ++kk) {
                    v2f a  = *reinterpret_cast<const v2f*>(xa + 4 * kk);
                    v2f bb = *reinterpret_cast<const v2f*>(ca + 4 * kk);
                    acc1 = __builtin_amdgcn_wmma_f32_16x16x4_f32(false, a, false, bb,
                                                                 (short)0, acc1, false, false);
                }
            }
            __syncthreads();
        }

        // D frag: VGPR v -> row m0 + v + 8*half, col n0 + lm
#pragma unroll
        for (int v = 0; v < 8; ++v)
            dotL[(m0a + v + 8 * half) * KPAD + (n0a + lm)] = acc0[v];
        if (tile1 < 14) {
#pragma unroll
            for (int v = 0; v < 8; ++v)
                dotL[(m0b + v + 8 * half) * KPAD + (n0b + lm)] = acc1[v];
        }
        if (t < KCL) cnt[t] = 0;
        __syncthreads();

        // --- argmin + counts ---
        if (t < NSEL) {
            float bd = 3.402823466e38f;
            int   bk = 0;
            for (int k = 0; k < KCL; ++k) {
                float d = cnorm[k] - 2.0f * dotL[t * KPAD + k];
                if (d < bd) { bd = d; bk = k; }
            }
            labels[t] = bk;
            atomicAdd(&cnt[bk], 1);
        }
        __syncthreads();

        if (it == ITERS) break;

        // --- center update: 4 channel chunks; thread owns channels 4t..4t+3 ---
        const int c0 = t * 4;
        for (int ch = 0; ch < 4; ++ch) {
            for (int j = t; j < KCL * 1024; j += 256) u.sums[j] = 0.0f;
            __syncthreads();
            for (int n = 0; n < NSEL; ++n) {
                int lab = labels[n];
                const float* xr = Xb + (size_t)n * CC_ + ch * 1024 + c0;
                float* sr = u.sums + lab * 1024 + c0;
                sr[0] += xr[0]; sr[1] += xr[1]; sr[2] += xr[2]; sr[3] += xr[3];
            }
            __syncthreads();
            for (int k = 0; k < KCL; ++k) {
                int cc = cnt[k];
                if (cc > 0) {                      // cnt==0 rows keep old centers
                    float inv = 1.0f / (float)cc;
                    float* cr = Cb + (size_t)k * CC_ + ch * 1024 + c0;
                    const float* sr = u.sums + k * 1024 + c0;
                    cr[0] = sr[0] * inv; cr[1] = sr[1] * inv;
                    cr[2] = sr[2] * inv; cr[3] = sr[3] * inv;
                }
            }
            __syncthreads();
        }
    }

    if (t < MPAD) labelsG[b * MPAD + t] = (t < NSEL) ? labels[t] : 0;
    if (t < KCL)  cntG[b * KCL + t] = cnt[t];
}

// ---------------------------------------------------------------------------
// Kernel 3: aggregate 400 tokens -> 20x1024 centers -> mean -> L2 normalize
// ---------------------------------------------------------------------------
__global__ void k_aggregate(const float* __restrict__ Xbuf, const int* __restrict__ labelsG,
                            const int* __restrict__ cntG, float* __restrict__ out)
{
    const int b = blockIdx.x;
    const int t = threadIdx.x;

    __shared__ float sums[KCL * 1024];
    __shared__ int   labels[NSEL];
    __shared__ float cw[KCL];
    __shared__ float red[256];

    if (t < NSEL) labels[t] = labelsG[b * MPAD + t];
    if (t < KCL) {
        int c4 = 4 * cntG[b * KCL + t];
        cw[t] = 1.0f / (float)(c4 > 1 ? c4 : 1);
    }
    for (int j = t; j < KCL * 1024; j += 256) sums[j] = 0.0f;
    __syncthreads();

    const float* Xb = Xbuf + (size_t)b * MPAD * CC_;
    const int c0 = t * 4;
    for (int l = 0; l < 4; ++l) {
        for (int n = 0; n < NSEL; ++n) {
            int lab = labels[n];
            const float* xr = Xb + (size_t)n * CC_ + l * 1024 + c0;
            float* sr = sums + lab * 1024 + c0;
            sr[0] += xr[0]; sr[1] += xr[1]; sr[2] += xr[2]; sr[3] += xr[3];
        }
    }
    __syncthreads();

    float o[4] = {0.f, 0.f, 0.f, 0.f};
    for (int k = 0; k < KCL; ++k) {
        float w = cw[k];
        const float* sr = sums + k * 1024 + c0;
        o[0] += sr[0] * w; o[1] += sr[1] * w; o[2] += sr[2] * w; o[3] += sr[3] * w;
    }
    float p = 0.f;
#pragma unroll
    for (int j = 0; j < 4; ++j) { o[j] *= (1.0f / (float)KCL); p += o[j] * o[j]; }

    red[t] = p;
    __syncthreads();
    for (int s = 128; s > 0; s >>= 1) {
        if (t < s) red[t] += red[t + s];
        __syncthreads();
    }
    float inv = 1.0f / fmaxf(sqrtf(red[0]), 1e-12f);
    float* ob = out + (size_t)b * C_ + c0;
    ob[0] = o[0] * inv; ob[1] = o[1] * inv; ob[2] = o[2] * inv; ob[3] = o[3] * inv;
}

// ---------------------------------------------------------------------------
extern "C" void kernel_launch(void* const* d_in, const int* in_sizes, int n_in,
                              void* d_out, int out_size, void* d_ws, size_t ws_size,
                              hipStream_t stream) {
    (void)in_sizes; (void)n_in; (void)out_size; (void)ws_size;

    // setup_inputs() order is interleaved: pt0, am0, pt1, am1, pt2, am2, pt3, am3
    const float* pt0 = (const float*)d_in[0];
    const float* am0 = (const float*)d_in[1];
    const float* pt1 = (const float*)d_in[2];
    const float* am1 = (const float*)d_in[3];
    const float* pt2 = (const float*)d_in[4];
    const float* am2 = (const float*)d_in[5];
    const float* pt3 = (const float*)d_in[6];
    const float* am3 = (const float*)d_in[7];

    float* Xbuf    = (float*)d_ws;                       // 8*112*4096 floats
    float* Cbuf    = Xbuf + (size_t)B_ * MPAD * CC_;     // 8*32*4096 floats
    int*   selIdx  = (int*)(Cbuf + (size_t)B_ * KPAD * CC_);
    int*   labelsG = selIdx + B_ * NSEL;
    int*   cntG    = labelsG + B_ * MPAD;

    k_select_gather<<<B_, 256, 0, stream>>>(pt0, pt1, pt2, pt3,
                                            am0, am1, am2, am3,
                                            Xbuf, Cbuf, selIdx);
    k_kmeans<<<B_, 256, 0, stream>>>(Xbuf, Cbuf, labelsG, cntG);
    k_aggregate<<<B_, 256, 0, stream>>>(Xbuf, labelsG, cntG, (float*)d_out);
}